// TargetNoise_62895501082799
// MI455X (gfx1250) — compile-verified
//
#include <hip/hip_runtime.h>

// out[b,t,d] = noise[b,t,d] * SIGMA * clamp((max(x_len[b]-t,0)-TTA_MIN)/(TTA_MAX-TTA_MIN), 0, 1)
// B=32, T=2048, D=512  ->  pure streaming kernel, HBM-bandwidth bound (~256 MiB @ 23.3 TB/s).
// x (d_in[0]) is shape-only and is never read.

typedef __attribute__((ext_vector_type(4))) float v4f;

#ifndef SIGMA
#define SIGMA 0.1f
#endif

static constexpr float kTtaMin   = 5.0f;
static constexpr float kInvRange = 1.0f / (60.0f - 5.0f);   // 1/55

static constexpr int kT        = 2048;           // rows per batch
static constexpr int kD4       = 512 / 4;        // float4s per row = 128 (power of two)
static constexpr int kThreads  = 256;
static constexpr int kVec      = 4;              // float4s per thread
static constexpr int kTile     = kThreads * kVec; // 1024 float4s per block

// Note: batch stride in float4 units = kT*kD4 = 262144, a multiple of kTile (1024),
// so a block's tile never crosses a batch boundary -> b is block-uniform -> s_load.

__global__ __launch_bounds__(kThreads) void TargetNoise_62895501082799_kernel(
    const int*  __restrict__ x_len,   // [B] (int32 per harness contract)
    const v4f*  __restrict__ noise,   // [B*T*D4]
    v4f*        __restrict__ out,     // [B*T*D4]
    int n4)                           // total float4 count
{
    const int tileBase = (int)(blockIdx.x * (unsigned)kTile);

    // Block-uniform batch index -> scalar load of the row length.
    const int   b  = tileBase >> 18;          // / (kT*kD4) = / 262144
    const float xl = (float)x_len[b];

    const int base = tileBase + (int)threadIdx.x;

#pragma unroll
    for (int k = 0; k < kVec; ++k) {
        const int i = base + k * kThreads;
        if (i >= n4) break;

        // t = (i / kD4) % kT  (all powers of two)
        const int   t  = (i >> 7) & (kT - 1);

        // scale = clamp((x_len[b] - t - TTA_MIN) * 1/55, 0, 1)
        // (the inner max(.,0) is absorbed: negative tta already clamps to 0)
        float s = (xl - (float)t - kTtaMin) * kInvRange;
        s = fminf(fmaxf(s, 0.0f), 1.0f);
        const float f = SIGMA * s;

        // Stream: consumed-once load, written-once store -> non-temporal both ways.
        v4f v = __builtin_nontemporal_load(&noise[i]);
        v4f r;
        r.x = v.x * f;
        r.y = v.y * f;
        r.z = v.z * f;
        r.w = v.w * f;
        __builtin_nontemporal_store(r, &out[i]);
    }
}

extern "C" void kernel_launch(void* const* d_in, const int* in_sizes, int n_in,
                              void* d_out, int out_size, void* d_ws, size_t ws_size,
                              hipStream_t stream) {
    // setup_inputs order: d_in[0]=x (unused), d_in[1]=x_len, d_in[2]=noise
    const int* x_len = (const int*)d_in[1];
    const v4f* noise = (const v4f*)d_in[2];
    v4f*       out   = (v4f*)d_out;

    const int n4     = out_size / 4;                 // 8,388,608 for 32x2048x512
    const int blocks = (n4 + kTile - 1) / kTile;     // 8192

    TargetNoise_62895501082799_kernel<<<blocks, kThreads, 0, stream>>>(
        x_len, noise, out, n4);
}